// GraphSAGE_71468255805600
// MI455X (gfx1250) — compile-verified
//
#include <hip/hip_runtime.h>

#define DIM 128
#define NEG_SLOPE 0.01f

typedef __attribute__((ext_vector_type(2))) float v2f;
typedef __attribute__((ext_vector_type(8))) float v8f;

// ---------------------------------------------------------------- CSR build
__global__ void zero_int_kernel(int* __restrict__ p, int n) {
  int i = blockIdx.x * blockDim.x + threadIdx.x;
  if (i < n) p[i] = 0;
}

__global__ void count_kernel(const int* __restrict__ dst, int* __restrict__ cnt, int nE) {
  int e = blockIdx.x * blockDim.x + threadIdx.x;
  if (e < nE) atomicAdd(&cnt[dst[e]], 1);
}

// Single-block exclusive scan: wave32 shuffle scan + cross-wave scan + carry.
__global__ __launch_bounds__(1024) void scan_kernel(const int* __restrict__ cnt,
                                                    int* __restrict__ rowptr, int n) {
  __shared__ int wsum[32];
  __shared__ int woff[32];
  __shared__ int carry;
  const int tid  = threadIdx.x;
  const int lane = tid & 31;
  const int wave = tid >> 5;
  if (tid == 0) carry = 0;
  __syncthreads();
  for (int base = 0; base < n; base += 1024) {
    int i = base + tid;
    int v = (i < n) ? cnt[i] : 0;
    int x = v;
#pragma unroll
    for (int off = 1; off < 32; off <<= 1) {
      int y = __shfl_up(x, off, 32);
      if (lane >= off) x += y;
    }
    if (lane == 31) wsum[wave] = x;
    __syncthreads();
    if (wave == 0) {
      int w  = wsum[lane];
      int xs = w;
#pragma unroll
      for (int off = 1; off < 32; off <<= 1) {
        int y = __shfl_up(xs, off, 32);
        if (lane >= off) xs += y;
      }
      woff[lane] = xs - w;              // exclusive wave offsets
      if (lane == 31) wsum[31] = xs;    // chunk total
    }
    __syncthreads();
    int excl = x - v + woff[wave];
    if (i < n) rowptr[i] = carry + excl;
    __syncthreads();                    // all carry reads done
    if (tid == 0) carry += wsum[31];
    // barrier at top of next chunk (wsum write sync) orders the carry update
  }
  if (threadIdx.x == 0) rowptr[n] = carry;
}

__global__ void copy_int_kernel(const int* __restrict__ a, int* __restrict__ b, int n) {
  int i = blockIdx.x * blockDim.x + threadIdx.x;
  if (i < n) b[i] = a[i];
}

__global__ void fill_kernel(const int* __restrict__ src, const int* __restrict__ dst,
                            int* __restrict__ cursor, int* __restrict__ esrc, int nE) {
  int e = blockIdx.x * blockDim.x + threadIdx.x;
  if (e >= nE) return;
  int pos = atomicAdd(&cursor[dst[e]], 1);
  esrc[pos] = src[e];
}

// ------------------------------------------------- fused gather + normalize
// One wave per node; lane owns 4 contiguous floats of the 128-dim feature.
// out[v] = (h[v] + sum_{u->v} h[u]) / (in_deg(v) + 1)
__global__ __launch_bounds__(256) void aggregate_kernel(
    const float* __restrict__ h, const int* __restrict__ rowptr,
    const int* __restrict__ esrc, float* __restrict__ out, int n) {
  int v = blockIdx.x * (blockDim.x >> 5) + (threadIdx.x >> 5);
  if (v >= n) return;
  const int lane = threadIdx.x & 31;
  const int c = lane * 4;
  const int beg = rowptr[v];
  const int end = rowptr[v + 1];
  float4 acc = *(const float4*)(h + (size_t)v * DIM + c);   // self term
  int j = beg;
  for (; j + 1 < end; j += 2) {                             // 2-deep to hide latency
    int s0 = esrc[j];
    int s1 = esrc[j + 1];
    float4 t0 = *(const float4*)(h + (size_t)s0 * DIM + c);
    float4 t1 = *(const float4*)(h + (size_t)s1 * DIM + c);
    acc.x += t0.x; acc.y += t0.y; acc.z += t0.z; acc.w += t0.w;
    acc.x += t1.x; acc.y += t1.y; acc.z += t1.z; acc.w += t1.w;
  }
  if (j < end) {
    int s = esrc[j];
    float4 t = *(const float4*)(h + (size_t)s * DIM + c);
    acc.x += t.x; acc.y += t.y; acc.z += t.z; acc.w += t.w;
  }
  const float inv = 1.0f / (float)(end - beg + 1);
  acc.x *= inv; acc.y *= inv; acc.z *= inv; acc.w *= inv;
  *(float4*)(out + (size_t)v * DIM + c) = acc;
}

// ---------------------------------------------------------------- WMMA GEMM
__device__ __forceinline__ v8f wmma_f32_4(v2f a, v2f b, v8f c) {
#if defined(__HIP_DEVICE_COMPILE__) && __has_builtin(__builtin_amdgcn_wmma_f32_16x16x4_f32)
  return __builtin_amdgcn_wmma_f32_16x16x4_f32(
      /*neg_a=*/false, a, /*neg_b=*/false, b,
      /*c_mod=*/(short)0, c, /*reuse_a=*/false, /*reuse_b=*/false);
#else
  (void)a; (void)b;
  return c;   // host pass only
#endif
}

// Out[row0:row0+16, :] = act(A[row0:row0+16, :] @ W + bias)
// 256 threads = 8 waves; wave w owns output cols [16w, 16w+16). n % 16 == 0.
#define AS_LD 132   // 128 + 4 pad: float4-aligned, no LDS bank conflicts
__global__ __launch_bounds__(256) void gemm_bias_act_kernel(
    const float* __restrict__ A, const float* __restrict__ W,
    const float* __restrict__ bias, float* __restrict__ Out, int applyAct) {
  __shared__ float As[16 * AS_LD];
  const int row0 = blockIdx.x * 16;
  const int t = threadIdx.x;
  {
    int r = t >> 4;            // 0..15
    int c = (t & 15) * 8;      // 0..120
    const float4* g = (const float4*)(A + (size_t)(row0 + r) * DIM + c);
    float4 v0 = g[0];
    float4 v1 = g[1];
    *(float4*)&As[r * AS_LD + c]     = v0;
    *(float4*)&As[r * AS_LD + c + 4] = v1;
  }
  __syncthreads();

  const int wave  = t >> 5;
  const int lane  = t & 31;
  const int col0  = wave * 16;
  const int mn    = lane & 15;          // M for A frag, N for B/C frags
  const int khalf = (lane >> 4) << 1;   // lanes 0-15 -> K{0,1}; 16-31 -> K{2,3}

  v8f acc = {};
#pragma unroll 4
  for (int k0 = 0; k0 < DIM; k0 += 4) {
    v2f a, b;
    a.x = As[mn * AS_LD + k0 + khalf];
    a.y = As[mn * AS_LD + k0 + khalf + 1];
    b.x = W[(size_t)(k0 + khalf) * DIM + col0 + mn];
    b.y = W[(size_t)(k0 + khalf + 1) * DIM + col0 + mn];
    acc = wmma_f32_4(a, b, acc);
  }

  const float bv = bias[col0 + mn];
  const int mbase = (lane >> 4) << 3;   // C/D: lanes 0-15 VGPR r->M=r; 16-31 -> M=8+r
#pragma unroll
  for (int r = 0; r < 8; ++r) {
    float vv = acc[r] + bv;
    if (applyAct) vv = (vv >= 0.0f) ? vv : NEG_SLOPE * vv;
    Out[(size_t)(row0 + mbase + r) * DIM + col0 + mn] = vv;
  }
}

// ---------------------------------------------------------------- launcher
extern "C" void kernel_launch(void* const* d_in, const int* in_sizes, int n_in,
                              void* d_out, int out_size, void* d_ws, size_t ws_size,
                              hipStream_t stream) {
  const float* x   = (const float*)d_in[0];
  const int*   src = (const int*)d_in[1];
  const int*   dst = (const int*)d_in[2];
  const float* W1  = (const float*)d_in[3];
  const float* b1  = (const float*)d_in[4];
  const float* W2  = (const float*)d_in[5];
  const float* b2  = (const float*)d_in[6];
  float* out = (float*)d_out;

  const int n  = in_sizes[0] / DIM;   // 50000
  const int nE = in_sizes[1];         // 600000

  // workspace: hn [n*DIM f32] | h1 [n*DIM f32] | rowptr [n+1] | cursor [n] | esrc [nE]
  float* hn     = (float*)d_ws;
  float* h1     = hn + (size_t)n * DIM;
  int*   rowptr = (int*)(h1 + (size_t)n * DIM);
  int*   cursor = rowptr + (n + 1);
  int*   esrc   = cursor + n;

  const int T = 256;
  const int aggBlocks  = (n + 7) / 8;   // 8 waves/block, 1 node/wave
  const int gemmBlocks = n / 16;        // 3125 exact

  // ---- CSR build (graph shared by both layers) ----
  zero_int_kernel<<<(n + T - 1) / T, T, 0, stream>>>(cursor, n);          // counts in cursor
  count_kernel<<<(nE + T - 1) / T, T, 0, stream>>>(dst, cursor, nE);
  scan_kernel<<<1, 1024, 0, stream>>>(cursor, rowptr, n);
  copy_int_kernel<<<(n + T - 1) / T, T, 0, stream>>>(rowptr, cursor, n);  // fill cursors
  fill_kernel<<<(nE + T - 1) / T, T, 0, stream>>>(src, dst, cursor, esrc, nE);

  // ---- layer 1 ----
  aggregate_kernel<<<aggBlocks, T, 0, stream>>>(x, rowptr, esrc, hn, n);
  gemm_bias_act_kernel<<<gemmBlocks, T, 0, stream>>>(hn, W1, b1, h1, 1);

  // ---- layer 2 ----
  aggregate_kernel<<<aggBlocks, T, 0, stream>>>(h1, rowptr, esrc, hn, n);
  gemm_bias_act_kernel<<<gemmBlocks, T, 0, stream>>>(hn, W2, b2, out, 0);
}